// GCN_33500744909168
// MI455X (gfx1250) — compile-verified
//
#include <hip/hip_runtime.h>

#define NN 100000
#define NE 1600000
#define DIMF 128
#define LDS_STRIDE 132   // 128 + 4 pad floats: kills 16-way LDS bank conflicts, keeps 16B alignment

typedef __attribute__((ext_vector_type(2))) float v2f;
typedef __attribute__((ext_vector_type(8))) float v8f;

// ---------------- degree / normalization ----------------

__global__ void zero_c_kernel(float* __restrict__ c_src, float* __restrict__ c_dst) {
    int i = blockIdx.x * 256 + threadIdx.x;
    if (i < NN) { c_src[i] = 0.0f; c_dst[i] = 0.0f; }
}

__global__ void degree_kernel(const int* __restrict__ src, const int* __restrict__ dst,
                              float* __restrict__ deg_out, float* __restrict__ deg_in) {
    int e = blockIdx.x * 256 + threadIdx.x;
    if (e < NE) {
        atomicAdd(&deg_out[src[e]], 1.0f);
        atomicAdd(&deg_in[dst[e]], 1.0f);
    }
}

__global__ void finalize_c_kernel(float* __restrict__ c_src, float* __restrict__ c_dst) {
    int i = blockIdx.x * 256 + threadIdx.x;
    if (i < NN) {
        // +1.0f accounts for the self-loop on every node; deg+1 >= 1 so max() is implicit
        c_src[i] = rsqrtf(c_src[i] + 1.0f);
        c_dst[i] = rsqrtf(c_dst[i] + 1.0f);
    }
}

// ---------------- per-layer aggregation ----------------

// agg = x * c_src  (this is exactly the self-loop contribution h[i] -> agg[i])
__global__ void init_agg_kernel(const float4* __restrict__ x, const float* __restrict__ c_src,
                                float4* __restrict__ agg) {
    int t = blockIdx.x * 256 + threadIdx.x;     // float4 granularity: NN*32 slots
    if (t < NN * 32) {
        int node = t >> 5;
        float s = c_src[node];
        float4 v = x[t];
        agg[t] = make_float4(v.x * s, v.y * s, v.z * s, v.w * s);
    }
}

// one wave32 per edge; each lane moves a float4 (4 atomic f32 adds, no return -> STOREcnt)
__global__ __launch_bounds__(256)
void scatter_kernel(const float4* __restrict__ x, const int* __restrict__ src,
                    const int* __restrict__ dst, const float* __restrict__ c_src,
                    float* __restrict__ agg) {
    int wave = threadIdx.x >> 5;
    int lane = threadIdx.x & 31;
    int e = blockIdx.x * 8 + wave;
    if (e >= NE) return;
    int s = src[e];
    int d = dst[e];
    float cs = c_src[s];
    float4 v = x[(size_t)s * 32 + lane];
    float* p = agg + (size_t)d * DIMF + lane * 4;
    atomicAdd(p + 0, v.x * cs);
    atomicAdd(p + 1, v.y * cs);
    atomicAdd(p + 2, v.z * cs);
    atomicAdd(p + 3, v.w * cs);
}

// ---------------- WMMA GEMM: out = relu?((agg * c_dst) @ W + b) ----------------
// block: 256 threads = 8 waves; block owns 16 nodes (M=16), wave w owns N-tile [16w,16w+16)
// K = 128 -> 32 x V_WMMA_F32_16X16X4_F32 per wave (exact f32 math).

__global__ __launch_bounds__(256)
void gemm_wmma_kernel(const float* __restrict__ agg, const float* __restrict__ c_dst,
                      const float* __restrict__ W, const float* __restrict__ bias,
                      float* __restrict__ out, int do_relu) {
    __shared__ float Atile[16 * LDS_STRIDE];    // padded 16x128 c_dst-scaled A tile (~8.4 KB)

    const int node0 = blockIdx.x * 16;
    const int t = threadIdx.x;

    // cooperative load + row scaling: 512 float4 slots (32 per row), 2 per thread
    {
        const float4* srcp = (const float4*)(agg + (size_t)node0 * DIMF);
#pragma unroll
        for (int i = 0; i < 2; ++i) {
            int idx = t + i * 256;              // 0..511
            int row = idx >> 5;                 // node within tile
            int col4 = idx & 31;                // float4 column
            float s = c_dst[node0 + row];
            float4 v = srcp[idx];
            float4* dstp = (float4*)&Atile[row * LDS_STRIDE + col4 * 4];  // stride%4==0 -> 16B aligned
            *dstp = make_float4(v.x * s, v.y * s, v.z * s, v.w * s);
        }
    }
    __syncthreads();

    const int wave = t >> 5;
    const int lane = t & 31;
    const int n0 = wave * 16;
    const int l15 = lane & 15;                  // A: M index; B: N index
    const int kh  = (lane >> 4) * 2;            // K sub-offset per lane half

    v8f acc = {};
#pragma unroll
    for (int k0 = 0; k0 < DIMF; k0 += 4) {
        int kb = k0 + kh;                       // even -> 8B-aligned float2 LDS read
        float2 av = *(const float2*)&Atile[l15 * LDS_STRIDE + kb];
        v2f a; a.x = av.x; a.y = av.y;
        v2f b;
        b.x = W[(size_t)kb * DIMF + n0 + l15];
        b.y = W[(size_t)(kb + 1) * DIMF + n0 + l15];
        // 8 args: (neg_a, A, neg_b, B, c_mod, C, reuse_a, reuse_b)
        acc = __builtin_amdgcn_wmma_f32_16x16x4_f32(
            false, a, false, b, (short)0, acc, false, false);
    }

    // C/D layout: VGPR r -> M = r + 8*(lane>>4), N = lane&15
    const int n = n0 + l15;
    const float bn = bias[n];
    const int mbase = (lane >> 4) * 8;
#pragma unroll
    for (int r = 0; r < 8; ++r) {
        float v = acc[r] + bn;
        if (do_relu) v = fmaxf(v, 0.0f);
        out[(size_t)(node0 + r + mbase) * DIMF + n] = v;
    }
}

// ---------------- driver ----------------

extern "C" void kernel_launch(void* const* d_in, const int* in_sizes, int n_in,
                              void* d_out, int out_size, void* d_ws, size_t ws_size,
                              hipStream_t stream) {
    const float* node_attr = (const float*)d_in[0];
    const float* W0 = (const float*)d_in[1];
    const float* b0 = (const float*)d_in[2];
    const float* W1 = (const float*)d_in[3];
    const float* b1 = (const float*)d_in[4];
    const float* W2 = (const float*)d_in[5];
    const float* b2 = (const float*)d_in[6];
    const int* edge = (const int*)d_in[7];
    const int* src = edge;            // edge_index[0]
    const int* dst = edge + NE;       // edge_index[1]
    float* out = (float*)d_out;

    char* ws = (char*)d_ws;
    float* c_src = (float*)ws;                                    // NN floats
    float* c_dst = c_src + NN;                                    // NN floats
    float* agg   = (float*)(ws + (size_t)2 * NN * sizeof(float)); // NN*128 floats (800000 % 16 == 0)

    const int nblk_n = (NN + 255) / 256;
    const int nblk_e = (NE + 255) / 256;
    const int nblk_s = (NE + 7) / 8;       // 8 waves (edges) per 256-thread block
    const int nblk_i = (NN * 32 + 255) / 256;
    const int nblk_g = NN / 16;            // 100000 / 16 = 6250 exactly

    // normalization coefficients
    zero_c_kernel<<<nblk_n, 256, 0, stream>>>(c_src, c_dst);
    degree_kernel<<<nblk_e, 256, 0, stream>>>(src, dst, c_src, c_dst);
    finalize_c_kernel<<<nblk_n, 256, 0, stream>>>(c_src, c_dst);

    // layer 0: node_attr -> out (relu)
    init_agg_kernel<<<nblk_i, 256, 0, stream>>>((const float4*)node_attr, c_src, (float4*)agg);
    scatter_kernel<<<nblk_s, 256, 0, stream>>>((const float4*)node_attr, src, dst, c_src, agg);
    gemm_wmma_kernel<<<nblk_g, 256, 0, stream>>>(agg, c_dst, W0, b0, out, 1);

    // layer 1: out -> out (relu); scatter fully consumes `out` before gemm rewrites it
    init_agg_kernel<<<nblk_i, 256, 0, stream>>>((const float4*)out, c_src, (float4*)agg);
    scatter_kernel<<<nblk_s, 256, 0, stream>>>((const float4*)out, src, dst, c_src, agg);
    gemm_wmma_kernel<<<nblk_g, 256, 0, stream>>>(agg, c_dst, W1, b1, out, 1);

    // layer 2: out -> out (no relu)
    init_agg_kernel<<<nblk_i, 256, 0, stream>>>((const float4*)out, c_src, (float4*)agg);
    scatter_kernel<<<nblk_s, 256, 0, stream>>>((const float4*)out, src, dst, c_src, agg);
    gemm_wmma_kernel<<<nblk_g, 256, 0, stream>>>(agg, c_dst, W2, b2, out, 0);
}